// UpSample_34608846471731
// MI455X (gfx1250) — compile-verified
//
#include <hip/hip_runtime.h>

typedef __attribute__((ext_vector_type(2))) float v2f;
typedef __attribute__((ext_vector_type(4))) float v4f;
typedef __attribute__((ext_vector_type(8))) float v8f;

#define B_SZ   4
#define N_PTS  65536
#define F_CH   128
#define UNITS  128
#define K_DIM  256           // 2*F
#define LDS_STRIDE 260       // 260 % 64 == 4 -> conflict-free b64 reads across wave32

// ---------------- kernel 1: zero the scatter workspace ----------------
__global__ void zero_ws(float* __restrict__ p) {
    unsigned i = blockIdx.x * blockDim.x + threadIdx.x;   // grid sized exactly: B*N*F/4 vec4 elems
    v4f z = {0.f, 0.f, 0.f, 0.f};
    ((v4f*)p)[i] = z;
}

// ---------------- kernel 2: gather + scatter-add (atomic f32) ----------------
__global__ void scatter_add(const float* __restrict__ feats,
                            const int*   __restrict__ iidx,
                            const int*   __restrict__ uidx,
                            float*       __restrict__ scat) {
    unsigned e = blockIdx.x * blockDim.x + threadIdx.x;   // one per (b, i, f); B*N*F total
    unsigned f = e & (F_CH - 1);
    unsigned i = (e >> 7) & (N_PTS - 1);
    unsigned b = e >> 23;                                  // 7 + 16 bits
    const unsigned half = N_PTS / 2;

    unsigned srcRow;
    if (i < half) {
        srcRow = b * half + i;                             // first half: feats as-is
    } else {
        srcRow = b * half + (unsigned)iidx[b * half + (i - half)];  // gathered copy
    }
    float v = feats[(size_t)srcRow * F_CH + f];

    unsigned dst = (unsigned)uidx[b * N_PTS + i];
    atomicAdd(&scat[((size_t)b * N_PTS + dst) * F_CH + f], v);
}

// ---------------- kernel 3: fused concat + GEMM(256x128) + bias + ReLU ----------------
// Block: 256 threads = 8 waves. Wave handles 16 rows x 128 units via V_WMMA_F32_16X16X4_F32.
__global__ void __launch_bounds__(256)
fused_gemm(const float* __restrict__ res,
           const float* __restrict__ scat,
           const float* __restrict__ W,      // (256,128) row-major
           const float* __restrict__ bias,   // (128,)
           float*       __restrict__ out) {  // (B*N,128)
    extern __shared__ float Wt[];            // [UNITS][LDS_STRIDE], transposed W: Wt[n][k]

    const int tid = threadIdx.x;

    // Stage W transposed into LDS; k is the fast index -> conflict-free LDS writes.
    for (int idx = tid; idx < UNITS * K_DIM; idx += 256) {
        int k = idx & (K_DIM - 1);
        int n = idx >> 8;
        Wt[n * LDS_STRIDE + k] = W[k * UNITS + n];
    }
    __syncthreads();

    const int wave  = tid >> 5;
    const int lane  = tid & 31;
    const int lrow  = lane & 15;             // M (for A) or N column (for B/C)
    const int kbase = (lane >> 4) << 1;      // 0 for lanes 0-15, 2 for lanes 16-31

    const long long rowbase = (long long)blockIdx.x * 128 + wave * 16;
    const float* arow0 = res  + (rowbase + lrow) * (long long)F_CH;  // K 0..127
    const float* arow1 = scat + (rowbase + lrow) * (long long)F_CH;  // K 128..255

    v8f acc[8];
#pragma unroll
    for (int t = 0; t < 8; ++t) acc[t] = (v8f){0.f,0.f,0.f,0.f,0.f,0.f,0.f,0.f};

    // K = 0..127 from res
#pragma unroll 4
    for (int kk = 0; kk < 32; ++kk) {
        const int k = (kk << 2) + kbase;
        v2f a = *(const v2f*)(arow0 + k);
#pragma unroll
        for (int nt = 0; nt < 8; ++nt) {
            const int n = (nt << 4) + lrow;
            v2f bv = *(const v2f*)&Wt[n * LDS_STRIDE + k];
            acc[nt] = __builtin_amdgcn_wmma_f32_16x16x4_f32(
                false, a, false, bv, (short)0, acc[nt], false, false);
        }
    }
    // K = 128..255 from scattered
#pragma unroll 4
    for (int kk = 0; kk < 32; ++kk) {
        const int k = (kk << 2) + kbase;             // local k in [0,128)
        v2f a = *(const v2f*)(arow1 + k);
#pragma unroll
        for (int nt = 0; nt < 8; ++nt) {
            const int n = (nt << 4) + lrow;
            v2f bv = *(const v2f*)&Wt[n * LDS_STRIDE + (k + 128)];
            acc[nt] = __builtin_amdgcn_wmma_f32_16x16x4_f32(
                false, a, false, bv, (short)0, acc[nt], false, false);
        }
    }

    // Epilogue: bias + ReLU + store. C layout: VGPR r, lanes0-15 -> M=r, lanes16-31 -> M=r+8.
    const int mhi = (lane >> 4) << 3;
#pragma unroll
    for (int nt = 0; nt < 8; ++nt) {
        const int col = (nt << 4) + lrow;
        const float bb = bias[col];
#pragma unroll
        for (int r = 0; r < 8; ++r) {
            float v = acc[nt][r] + bb;
            v = v > 0.f ? v : 0.f;
            out[(rowbase + r + mhi) * (long long)UNITS + col] = v;
        }
    }
}

extern "C" void kernel_launch(void* const* d_in, const int* in_sizes, int n_in,
                              void* d_out, int out_size, void* d_ws, size_t ws_size,
                              hipStream_t stream) {
    (void)in_sizes; (void)n_in; (void)out_size; (void)ws_size;
    const float* feats = (const float*)d_in[0];   // (4, 32768, 128)
    const int*   iidx  = (const int*)  d_in[1];   // (4, 32768)
    const int*   uidx  = (const int*)  d_in[2];   // (4, 65536)
    const float* res   = (const float*)d_in[3];   // (4, 65536, 128)
    const float* W     = (const float*)d_in[4];   // (256, 128)
    const float* bias  = (const float*)d_in[5];   // (128,)
    float* out  = (float*)d_out;                  // (4, 65536, 128)
    float* scat = (float*)d_ws;                   // (4, 65536, 128) scratch

    const int total   = B_SZ * N_PTS * F_CH;      // 33,554,432
    zero_ws    <<<total / 4 / 256, 256, 0, stream>>>(scat);
    scatter_add<<<total / 256,     256, 0, stream>>>(feats, iidx, uidx, scat);

    const int rows = B_SZ * N_PTS;                // 262,144
    const size_t lds_bytes = (size_t)UNITS * LDS_STRIDE * sizeof(float);  // 133,120 B
    fused_gemm <<<rows / 128, 256, lds_bytes, stream>>>(res, scat, W, bias, out);
}